// Fastsum_2362232012840
// MI455X (gfx1250) — compile-verified
//
#include <hip/hip_runtime.h>
#include <cmath>

#define TWO_PI_F 6.283185307179586f
#define PI_F     3.14159265358979f
#define NFT      1024
#define NSEG     32

typedef float v2f __attribute__((ext_vector_type(2)));
typedef float v8f __attribute__((ext_vector_type(8)));

__device__ __forceinline__ v8f zero8() {
    v8f z = {0.f,0.f,0.f,0.f,0.f,0.f,0.f,0.f};
    return z;
}

// D = A(16x4) * B(4x16) + C, fp32 WMMA (CDNA5)
__device__ __forceinline__ v8f wmma4(v2f a, v2f b, v8f c) {
    return __builtin_amdgcn_wmma_f32_16x16x4_f32(false, a, false, b, (short)0, c,
                                                 false, false);
}

// cos/sin of (2*pi*turns), explicit range reduction keeps HW trig in-domain.
__device__ __forceinline__ void sincos_turns(float t, float* c, float* s) {
    t -= rintf(t);
    float th = TWO_PI_F * t;
    *c = __cosf(th);
    *s = __sinf(th);
}

__device__ __forceinline__ void sincos_turns2(v2f t, v2f* c, v2f* s) {
    float c0, s0, c1, s1;
    sincos_turns(t.x, &c0, &s0);
    sincos_turns(t.y, &c1, &s1);
    v2f cv = { c0, c1 }, sv = { s0, s1 };
    *c = cv; *s = sv;
}

// packed complex multiply (ar+i*ai) *= (br+i*bi), elementwise on 2 points
__device__ __forceinline__ void cmul2(v2f& ar, v2f& ai, v2f br, v2f bi) {
    v2f tr = ar * br - ai * bi;
    ai     = ar * bi + ai * br;
    ar     = tr;
}

// ---------------- pass 1: max row norm^2 over x and y -----------------------
__global__ void k_maxnorm(const float* __restrict__ x, const float* __restrict__ y,
                          int N, int M, int d, unsigned int* maxbits) {
    int i = blockIdx.x * blockDim.x + threadIdx.x;
    if (i >= N + M) return;
    const float* row = (i < N) ? (x + (size_t)i * d) : (y + (size_t)(i - N) * d);
    float s = 0.f;
    for (int c = 0; c < d; c += 4) {
        float4 v = *(const float4*)(row + c);
        s += v.x * v.x + v.y * v.y + v.z * v.z + v.w * v.w;
    }
    atomicMax(maxbits, __float_as_uint(s));  // s >= 0: float order == uint order
}

// ---------------- pass 2: projections px = factor*(x . xi), py likewise -----
__global__ void k_proj(const float* __restrict__ x, const float* __restrict__ y,
                       const float* __restrict__ xis, const unsigned int* maxbits,
                       int N, int M, int d, int P,
                       float* __restrict__ px, float* __restrict__ py) {
    int i = blockIdx.x * blockDim.x + threadIdx.x;
    if (i >= P * (N + M)) return;
    float factor = 0.3f * rsqrtf(__uint_as_float(*maxbits));
    int p, j; const float* row; float* dst;
    if (i < P * N) { p = i / N; j = i % N; row = x + (size_t)j * d; dst = px + (size_t)p * N + j; }
    else { int t = i - P * N; p = t / M; j = t % M; row = y + (size_t)j * d; dst = py + (size_t)p * M + j; }
    const float* xi = xis + (size_t)p * d;
    float s = 0.f;
    for (int c = 0; c < d; c += 4) {
        float4 a = *(const float4*)(row + c);
        float4 b = *(const float4*)(xi + c);
        s += a.x * b.x + a.y * b.y + a.z * b.z + a.w * b.w;
    }
    *dst = factor * s;
}

// ---------------- pass 3: Gaussian slice kernel FT (log-space) --------------
__global__ void k_kft(const float* __restrict__ scale, const unsigned int* maxbits,
                      int d, float lgamma_half_d, float* __restrict__ kft) {
    int t = blockIdx.x * blockDim.x + threadIdx.x;
    if (t >= NFT) return;
    float factor = 0.3f * rsqrtf(__uint_as_float(*maxbits));
    float sigma = scale[0] * factor;
    float s2 = sigma * sigma;
    float df = (float)d;
    float log_c = 0.5f * df * logf(PI_F) - lgamma_half_d + 0.5f * df * logf(TWO_PI_F * s2);
    float k = (float)(t - NFT / 2);
    float v = 0.f;
    if (k != 0.f)
        v = expf(log_c + (df - 1.f) * logf(fabsf(k)) - 2.f * PI_F * PI_F * s2 * k * k);
    kft[t] = v;
}

// ---------------- pass 4: adjoint NDFT as complex WMMA matmul ---------------
// Xhat[a,b] = sum_j (w_j * z_j^(b-512)) * u_j^a,  u = z^16, z = exp(i*2pi*px)
// A-tile phases for t=0..3 advance by exp(i*2pi*256*p): rotation recurrence.
__global__ void __launch_bounds__(32)
k_adjoint(const float* __restrict__ px, const float* __restrict__ w,
          int N, float* __restrict__ xhat) {
    int p = blockIdx.y;
    int seg = blockIdx.x;
    int lane = threadIdx.x;
    int h = lane >> 4;      // which k-pair this lane supplies
    int m = lane & 15;      // A row (a within block) / B col (b)
    int jlen = N / NSEG;    // multiple of 4
    int j0base = seg * jlen;
    const float* pxp = px + (size_t)p * N;

    v8f Xr[4], Xi[4];
#pragma unroll
    for (int t = 0; t < 4; ++t) { Xr[t] = zero8(); Xi[t] = zero8(); }

    for (int j0 = j0base; j0 < j0base + jlen; j0 += 4) {
        int jA = j0 + 2 * h, jB = jA + 1;
        v2f pv = { pxp[jA], pxp[jB] };
        v2f wv = { w[jA],   w[jB]   };
        // B[k][b] = w_j * exp(i*2pi*(b-512)*p_j)
        v2f cB, sB;
        sincos_turns2(((float)m - 512.0f) * pv, &cB, &sB);
        v2f Br  = wv * cB;
        v2f Bi  = wv * sB;
        v2f BiN = -Bi;
        // A base (t=0): exp(i*2pi*16*m*p); step: exp(i*2pi*256*p)
        v2f Ar, Ai, Sr, Si;
        sincos_turns2((16.0f * (float)m) * pv, &Ar, &Ai);
        sincos_turns2(256.0f * pv, &Sr, &Si);
#pragma unroll
        for (int t = 0; t < 4; ++t) {
            Xr[t] = wmma4(Ar, Br,  Xr[t]);   // +ArBr
            Xr[t] = wmma4(Ai, BiN, Xr[t]);   // -AiBi
            Xi[t] = wmma4(Ar, Bi,  Xi[t]);   // +ArBi
            Xi[t] = wmma4(Ai, Br,  Xi[t]);   // +AiBr
            if (t < 3) cmul2(Ar, Ai, Sr, Si);
        }
    }
    // D[m][n]: vgpr r = row r (lanes 0-15) / row r+8 (lanes 16-31), col = lane&15
    float* xh = xhat + (size_t)p * (NFT * 2);
#pragma unroll
    for (int t = 0; t < 4; ++t)
#pragma unroll
        for (int r = 0; r < 8; ++r) {
            int a = 16 * t + r + 8 * h;
            int idx = (a * 16 + m) * 2;
            atomicAdd(&xh[idx + 0], Xr[t][r]);
            atomicAdd(&xh[idx + 1], Xi[t][r]);
        }
}

// ---------------- pass 5: g = kft * xhat ------------------------------------
__global__ void k_conv(const float* __restrict__ xhat, const float* __restrict__ kft,
                       int P, float* __restrict__ g) {
    int i = blockIdx.x * blockDim.x + threadIdx.x;
    if (i >= P * NFT) return;
    int k = i % NFT;            // storage index == k'; frequency = k'-512
    float f = kft[k];
    g[2 * i]     = f * xhat[2 * i];
    g[2 * i + 1] = f * xhat[2 * i + 1];
}

// ---------------- pass 6: forward NDFT as complex WMMA + epilogue -----------
// T[i,b] = sum_a g[a,b] * conj(u_i)^a ; out_i = (1/P) sum_p,b Re[e^{i2pi(512-b)py} T[i,b]]
// A-tile phases advance by exp(i*2pi*64*py) per chunk: rotation recurrence.
__global__ void __launch_bounds__(32)
k_forward(const float* __restrict__ py, const float* __restrict__ g,
          int M, int P, float* __restrict__ out) {
    int i0 = blockIdx.x * 16;
    int lane = threadIdx.x;
    int h = lane >> 4, m = lane & 15;
    float acc[8];
#pragma unroll
    for (int r = 0; r < 8; ++r) acc[r] = 0.f;

    for (int p = 0; p < P; ++p) {
        const float* pyp = py + (size_t)p * M;
        const float* gp  = g  + (size_t)p * (NFT * 2);
        float pv = pyp[i0 + m];                 // A-row's projection
        v8f Tr = zero8(), Ti = zero8();
        // unconjugated A for e = 2h, 2h+1; per-chunk step exp(i*2pi*64*pv)
        float c0, s0, c1, s1, cs, ss;
        sincos_turns(16.0f * (float)(2 * h) * pv, &c0, &s0);
        sincos_turns(16.0f * (float)(2 * h + 1) * pv, &c1, &s1);
        sincos_turns(64.0f * pv, &cs, &ss);
        v2f Ar = { c0, c1 }, Ai = { s0, s1 };
        v2f Sr = { cs, cs }, Si = { ss, ss };
#pragma unroll
        for (int c4 = 0; c4 < 16; ++c4) {
            int e0 = 4 * c4 + 2 * h, e1 = e0 + 1;
            int idx0 = (e0 * 16 + m) * 2, idx1 = (e1 * 16 + m) * 2;
            v2f Br = { gp[idx0],     gp[idx1]     };
            v2f Bi = { gp[idx0 + 1], gp[idx1 + 1] };
            v2f AiN = -Ai;
            // conj(A)*B: Tr += ArBr + AiBi ; Ti += ArBi - AiBr
            Tr = wmma4(Ar,  Br, Tr);
            Tr = wmma4(Ai,  Bi, Tr);
            Ti = wmma4(Ar,  Bi, Ti);
            Ti = wmma4(AiN, Br, Ti);
            if (c4 < 15) cmul2(Ar, Ai, Sr, Si);
        }
        // epilogue: multiply by e^{i*2pi*(512-b)*py_i} and take Re
#pragma unroll
        for (int r = 0; r < 8; ++r) {
            int irow = i0 + r + 8 * h;
            float pe = pyp[irow];
            float cc, sv;
            sincos_turns((512.0f - (float)m) * pe, &cc, &sv);
            acc[r] += cc * Tr[r] - sv * Ti[r];
        }
    }
    // reduce over b (16 lanes within each half-wave), write rows
#pragma unroll
    for (int r = 0; r < 8; ++r) {
        float v = acc[r];
        v += __shfl_xor(v, 1, 32);
        v += __shfl_xor(v, 2, 32);
        v += __shfl_xor(v, 4, 32);
        v += __shfl_xor(v, 8, 32);
        if (m == 0) out[i0 + r + 8 * h] = v / (float)P;
    }
}

extern "C" void kernel_launch(void* const* d_in, const int* in_sizes, int n_in,
                              void* d_out, int out_size, void* d_ws, size_t ws_size,
                              hipStream_t stream) {
    const float* x     = (const float*)d_in[0];
    const float* y     = (const float*)d_in[1];
    const float* w     = (const float*)d_in[2];
    const float* scale = (const float*)d_in[3];
    const float* xis   = (const float*)d_in[4];
    int N = in_sizes[2];            // 8192
    int d = in_sizes[0] / N;        // 32
    int M = in_sizes[1] / d;        // 8192
    int P = in_sizes[4] / d;        // 16
    float lgh = lgammaf(0.5f * (float)d);

    float* ws = (float*)d_ws;
    size_t o_kft = 16;
    size_t o_px  = o_kft + NFT;
    size_t o_py  = o_px + (size_t)P * N;
    size_t o_xh  = o_py + (size_t)P * M;
    size_t o_g   = o_xh + (size_t)P * NFT * 2;
    unsigned int* maxbits = (unsigned int*)ws;
    float* kft  = ws + o_kft;
    float* px   = ws + o_px;
    float* py   = ws + o_py;
    float* xhat = ws + o_xh;
    float* g    = ws + o_g;

    hipMemsetAsync(maxbits, 0, sizeof(unsigned int), stream);
    hipMemsetAsync(xhat, 0, (size_t)P * NFT * 2 * sizeof(float), stream);

    int t1 = N + M;
    k_maxnorm<<<(t1 + 255) / 256, 256, 0, stream>>>(x, y, N, M, d, maxbits);
    int t2 = P * (N + M);
    k_proj<<<(t2 + 255) / 256, 256, 0, stream>>>(x, y, xis, maxbits, N, M, d, P, px, py);
    k_kft<<<(NFT + 255) / 256, 256, 0, stream>>>(scale, maxbits, d, lgh, kft);
    dim3 gAdj(NSEG, P);
    k_adjoint<<<gAdj, 32, 0, stream>>>(px, w, N, xhat);
    k_conv<<<(P * NFT + 255) / 256, 256, 0, stream>>>(xhat, kft, P, g);
    k_forward<<<M / 16, 32, 0, stream>>>(py, g, M, P, (float*)d_out);
}